// ShapeGuidedRWKVBranch_15891378995282
// MI455X (gfx1250) — compile-verified
//
#include <hip/hip_runtime.h>

// ---------------------------------------------------------------------------
// RWKV-4 branch on gfx1250 (MI455X): fp32 WMMA GEMMs with async global->LDS
// double buffering (ASYNCcnt path) + chunked-parallel WKV scan.
// ---------------------------------------------------------------------------

typedef __attribute__((ext_vector_type(2))) float v2f;
typedef __attribute__((ext_vector_type(8))) float v8f;
typedef __attribute__((ext_vector_type(4))) int   v4i;

#define B_DIM  8
#define T_DIM  4096
#define C_DIM  768
#define M_DIM  (B_DIM * T_DIM)   // 32768 rows

#define BM 64
#define BN 128
#define BK 32
#define LDK 36                   // padded pitch: 144B = 16B aligned, conflict-free
#define NKB (C_DIM / BK)         // 24 K-blocks

#define NCH   8                  // WKV scan chunks per channel
#define CHLEN (T_DIM / NCH)      // 512 steps per chunk

// ----- gfx1250 async global->LDS (DMA, no VGPR staging), guarded ------------
#if defined(__HIP_DEVICE_COMPILE__) && \
    __has_builtin(__builtin_amdgcn_global_load_async_to_lds_b128)
#define HAVE_ASYNC_LDS 1
__device__ __forceinline__ void async_copy16(const float* g, float* l)
{
    __builtin_amdgcn_global_load_async_to_lds_b128(
        (v4i*)const_cast<float*>(g), (v4i*)l, 0, 0);
}
__device__ __forceinline__ void async_wait0()
{
#if __has_builtin(__builtin_amdgcn_s_wait_asynccnt)
    __builtin_amdgcn_s_wait_asynccnt(0);
#else
    asm volatile("s_wait_asynccnt 0" ::: "memory");
#endif
}
#else
#define HAVE_ASYNC_LDS 0
#endif

// ---------------------------------------------------------------------------
// Double-buffered NT-GEMM mainloop: out[m][n] = sum_k A[m][k] * W[n][k]
// 256 threads (8 waves) per 64(M) x 128(N) tile; wave w owns m-subtile w&3
// and n-half (w>>2)*64 -> 4x 16x16 fp32 WMMA accumulators.
// ---------------------------------------------------------------------------
__device__ __forceinline__ void gemm_mainloop(
    const float* __restrict__ A, const float* __restrict__ W,
    float (&As)[2][BM][LDK], float (&Bs)[2][BN][LDK],
    v8f (&acc)[4], int bm0, int bn0)
{
    const int tid  = threadIdx.x;
    const int lane = tid & 31;
    const int wave = tid >> 5;
    const int mt   = wave & 3;
    const int nb   = (wave >> 2) * 64;
    const int r    = lane & 15;
    const int kb2  = (lane >> 4) * 2;   // K sub-offset per f32 WMMA A/B layout

#pragma unroll
    for (int j = 0; j < 4; ++j) acc[j] = v8f{};

#if HAVE_ASYNC_LDS
    // ---- async DMA staging: global_load_async_to_lds_b128, ASYNCcnt -------
    auto stage_async = [&](int kb, int bsel) {
#pragma unroll
        for (int it = 0; it < 2; ++it) {          // A: 64x32 = 512 float4
            int idx = tid + it * 256;
            int row = idx >> 3;
            int c4  = (idx & 7) * 4;
            async_copy16(A + (size_t)(bm0 + row) * C_DIM + kb + c4,
                         &As[bsel][row][c4]);
        }
#pragma unroll
        for (int it = 0; it < 4; ++it) {          // B: 128x32 = 1024 float4
            int idx = tid + it * 256;
            int row = idx >> 3;
            int c4  = (idx & 7) * 4;
            async_copy16(W + (size_t)(bn0 + row) * C_DIM + kb + c4,
                         &Bs[bsel][row][c4]);
        }
    };
    stage_async(0, 0);
    async_wait0();
    __syncthreads();
#else
    // ---- fallback: register-staged copy ------------------------------------
    float4 ra[2], rb[4];
    auto load_regs = [&](int kb) {
#pragma unroll
        for (int it = 0; it < 2; ++it) {
            int idx = tid + it * 256;
            int row = idx >> 3;
            int c4  = (idx & 7) * 4;
            ra[it] = *(const float4*)(A + (size_t)(bm0 + row) * C_DIM + kb + c4);
        }
#pragma unroll
        for (int it = 0; it < 4; ++it) {
            int idx = tid + it * 256;
            int row = idx >> 3;
            int c4  = (idx & 7) * 4;
            rb[it] = *(const float4*)(W + (size_t)(bn0 + row) * C_DIM + kb + c4);
        }
    };
    auto store_lds = [&](int bsel) {
#pragma unroll
        for (int it = 0; it < 2; ++it) {
            int idx = tid + it * 256;
            int row = idx >> 3;
            int c4  = (idx & 7) * 4;
            *(float4*)(&As[bsel][row][c4]) = ra[it];
        }
#pragma unroll
        for (int it = 0; it < 4; ++it) {
            int idx = tid + it * 256;
            int row = idx >> 3;
            int c4  = (idx & 7) * 4;
            *(float4*)(&Bs[bsel][row][c4]) = rb[it];
        }
    };
    load_regs(0);
    store_lds(0);
    __syncthreads();
#endif

    int buf = 0;
    for (int kbi = 0; kbi < NKB; ++kbi) {
#if HAVE_ASYNC_LDS
        if (kbi + 1 < NKB) stage_async((kbi + 1) * BK, buf ^ 1);
#else
        if (kbi + 1 < NKB) load_regs((kbi + 1) * BK);
#endif

#pragma unroll
        for (int kk = 0; kk < BK; kk += 4) {
            v2f a = *(const v2f*)(&As[buf][mt * 16 + r][kk + kb2]);
            v2f bf[4];
#pragma unroll
            for (int j = 0; j < 4; ++j)           // hoist: 4 LDS loads in flight
                bf[j] = *(const v2f*)(&Bs[buf][nb + j * 16 + r][kk + kb2]);
#pragma unroll
            for (int j = 0; j < 4; ++j)
                acc[j] = __builtin_amdgcn_wmma_f32_16x16x4_f32(
                    false, a, false, bf[j], (short)0, acc[j], false, false);
        }

#if HAVE_ASYNC_LDS
        if (kbi + 1 < NKB) async_wait0();         // DMA into buf^1 landed
#else
        if (kbi + 1 < NKB) store_lds(buf ^ 1);
#endif
        __syncthreads();
        buf ^= 1;
    }
}

// ---------------------------------------------------------------------------
// Kernel 1: inverse permutation  inv[scan_indices[j]] = j
// ---------------------------------------------------------------------------
__global__ void build_inv(const int* __restrict__ sidx, int* __restrict__ inv)
{
    int j = blockIdx.x * blockDim.x + threadIdx.x;
    if (j < T_DIM) inv[sidx[j]] = j;
}

// ---------------------------------------------------------------------------
// Kernel 2: k/v/r GEMMs. grid.z selects weight; k,v rows scattered directly
// into scan order (row t -> inv[t]); r stored as sigmoid(r).
// ---------------------------------------------------------------------------
__global__ void gemm_kvr(const float* __restrict__ x,
                         const float* __restrict__ Wk,
                         const float* __restrict__ Wv,
                         const float* __restrict__ Wr,
                         float* __restrict__ k_s,
                         float* __restrict__ v_s,
                         float* __restrict__ sr,
                         const int* __restrict__ inv)
{
    __shared__ float As[2][BM][LDK];
    __shared__ float Bs[2][BN][LDK];

    const int zb = blockIdx.z;
    const float* W = (zb == 0) ? Wk : (zb == 1) ? Wv : Wr;
    const int bm0 = blockIdx.x * BM;
    const int bn0 = blockIdx.y * BN;

    v8f acc[4];
    gemm_mainloop(x, W, As, Bs, acc, bm0, bn0);

    const int tid  = threadIdx.x;
    const int lane = tid & 31;
    const int wave = tid >> 5;
    const int mt   = wave & 3;
    const int nb   = (wave >> 2) * 64;
    const int half = (lane >> 4) * 8;          // lanes 16-31 hold rows M+8
    const int gn_b = bn0 + nb + (lane & 15);

    float* kv = (zb == 0) ? k_s : v_s;

#pragma unroll
    for (int j = 0; j < 4; ++j) {
        int gn = gn_b + j * 16;
#pragma unroll
        for (int i = 0; i < 8; ++i) {
            int   gm  = bm0 + mt * 16 + half + i;
            float val = acc[j][i];
            if (zb == 2) {
                sr[(size_t)gm * C_DIM + gn] = 1.0f / (1.0f + __expf(-val));
            } else {
                int b = gm >> 12;              // / T_DIM
                int t = gm & (T_DIM - 1);
                size_t dst = ((size_t)(b << 12) + inv[t]) * C_DIM + gn;
                kv[dst] = val;
            }
        }
    }
}

// ---------------------------------------------------------------------------
// WKV chunked scan. State (aa,bb,pp) is linear in (A,B)=(aa*e^pp, bb*e^pp):
//   A' = e^w * A + e^k * v  per step, so chunk effects compose with the
//   stable merge  p = max(pp_in + L*w, pp_loc).
// Pass A: chunk-local states from zero init.
// Pass B: per-channel sequential combine -> incoming prefix per chunk.
// Pass C: replay each chunk from its prefix, emit y (scattered via sidx).
// ---------------------------------------------------------------------------
__global__ void wkv_chunk_local(const float* __restrict__ k_s,
                                const float* __restrict__ v_s,
                                const float* __restrict__ decay,
                                float* __restrict__ stA,
                                float* __restrict__ stB,
                                float* __restrict__ stP)
{
    int gid = blockIdx.x * blockDim.x + threadIdx.x;   // 0 .. B*NCH*C-1
    int c  = gid % C_DIM;
    int bq = gid / C_DIM;
    int q  = bq % NCH;
    int b  = bq / NCH;

    const float w = decay[c] * (1.0f / (float)T_DIM);

    const float* kp = k_s + (size_t)b * T_DIM * C_DIM + c;
    const float* vp = v_s + (size_t)b * T_DIM * C_DIM + c;
    const int t0 = q * CHLEN;

    float aa = 0.0f, bb = 0.0f, pp = -1e38f;
    for (int s = 0; s < CHLEN; ++s) {
        int t = t0 + s;
        __builtin_prefetch(kp + (size_t)(t + 16) * C_DIM, 0, 0);
        __builtin_prefetch(vp + (size_t)(t + 16) * C_DIM, 0, 0);
        float kt = kp[(size_t)t * C_DIM];
        float vt = vp[(size_t)t * C_DIM];
        float ww2 = w + pp;
        float p2  = fmaxf(ww2, kt);
        float e1  = __expf(ww2 - p2);
        float e2  = __expf(kt - p2);
        aa = e1 * aa + e2 * vt;
        bb = e1 * bb + e2;
        pp = p2;
    }
    stA[gid] = aa;
    stB[gid] = bb;
    stP[gid] = pp;
}

__global__ void wkv_combine(const float* __restrict__ decay,
                            float* __restrict__ stA,
                            float* __restrict__ stB,
                            float* __restrict__ stP)
{
    int ch = blockIdx.x * blockDim.x + threadIdx.x;    // 0 .. B*C-1
    int c  = ch % C_DIM;
    int b  = ch / C_DIM;

    const float w  = decay[c] * (1.0f / (float)T_DIM);
    const float Lw = (float)CHLEN * w;

    float aa = 0.0f, bb = 0.0f, pp = -1e38f;
#pragma unroll
    for (int q = 0; q < NCH; ++q) {
        size_t idx = ((size_t)b * NCH + q) * C_DIM + c;
        float la = stA[idx], lb = stB[idx], lp = stP[idx];
        // overwrite with incoming prefix (state before this chunk)
        stA[idx] = aa; stB[idx] = bb; stP[idx] = pp;
        // outgoing = merge(prefix shifted by L*w, chunk-local)
        float sh = pp + Lw;
        float p  = fmaxf(sh, lp);
        float ei = __expf(sh - p);
        float el = __expf(lp - p);
        aa = ei * aa + el * la;
        bb = ei * bb + el * lb;
        pp = p;
    }
}

__global__ void wkv_emit(const float* __restrict__ k_s,
                         const float* __restrict__ v_s,
                         const int*   __restrict__ sidx,
                         const float* __restrict__ decay,
                         const float* __restrict__ first,
                         const float* __restrict__ stA,
                         const float* __restrict__ stB,
                         const float* __restrict__ stP,
                         float* __restrict__ rwkv)
{
    int gid = blockIdx.x * blockDim.x + threadIdx.x;   // 0 .. B*NCH*C-1
    int c  = gid % C_DIM;
    int bq = gid / C_DIM;
    int q  = bq % NCH;
    int b  = bq / NCH;

    const float w = decay[c] * (1.0f / (float)T_DIM);
    const float u = first[c] * (1.0f / (float)T_DIM);

    const float* kp = k_s + (size_t)b * T_DIM * C_DIM + c;
    const float* vp = v_s + (size_t)b * T_DIM * C_DIM + c;
    float* yb       = rwkv + (size_t)b * T_DIM * C_DIM + c;
    const int t0 = q * CHLEN;

    float aa = stA[gid], bb = stB[gid], pp = stP[gid];

    for (int s = 0; s < CHLEN; ++s) {
        int t = t0 + s;
        __builtin_prefetch(kp + (size_t)(t + 16) * C_DIM, 0, 0);
        __builtin_prefetch(vp + (size_t)(t + 16) * C_DIM, 0, 0);
        float kt = kp[(size_t)t * C_DIM];
        float vt = vp[(size_t)t * C_DIM];

        float ww = u + kt;
        float p  = fmaxf(pp, ww);
        float e1 = __expf(pp - p);
        float e2 = __expf(ww - p);
        float y  = __fdividef(e1 * aa + e2 * vt, e1 * bb + e2);

        int row = sidx[t];                     // uniform across the wave
        yb[(size_t)row * C_DIM] = y;

        float ww2 = w + pp;
        float p2  = fmaxf(ww2, kt);
        float e1b = __expf(ww2 - p2);
        float e2b = __expf(kt - p2);
        aa = e1b * aa + e2b * vt;
        bb = e1b * bb + e2b;
        pp = p2;
    }
}

// ---------------------------------------------------------------------------
// LayerNorm over C + sigmoid(r) gate. One 256-thread block per row.
// ---------------------------------------------------------------------------
__global__ void ln_gate(const float* __restrict__ rwkv,
                        const float* __restrict__ sr,
                        const float* __restrict__ gamma,
                        const float* __restrict__ beta,
                        float* __restrict__ z)
{
    const int row = blockIdx.x;
    const float* xr = rwkv + (size_t)row * C_DIM;
    const int tid = threadIdx.x;

    float x0 = xr[tid];
    float x1 = xr[tid + 256];
    float x2 = xr[tid + 512];
    float s  = x0 + x1 + x2;
    float s2 = x0 * x0 + x1 * x1 + x2 * x2;

#pragma unroll
    for (int off = 16; off > 0; off >>= 1) {
        s  += __shfl_xor(s,  off, 32);
        s2 += __shfl_xor(s2, off, 32);
    }

    __shared__ float red[2][8];
    int wave = tid >> 5, lane = tid & 31;
    if (lane == 0) { red[0][wave] = s; red[1][wave] = s2; }
    __syncthreads();

    float ts = 0.0f, ts2 = 0.0f;
#pragma unroll
    for (int i = 0; i < 8; ++i) { ts += red[0][i]; ts2 += red[1][i]; }

    float mu   = ts * (1.0f / C_DIM);
    float var  = ts2 * (1.0f / C_DIM) - mu * mu;
    float rstd = rsqrtf(var + 1e-5f);

    const float* srr = sr + (size_t)row * C_DIM;
    float* zr        = z  + (size_t)row * C_DIM;
#pragma unroll
    for (int e = 0; e < 3; ++e) {
        int   c  = tid + e * 256;
        float xv = (e == 0) ? x0 : (e == 1) ? x1 : x2;
        zr[c] = ((xv - mu) * rstd * gamma[c] + beta[c]) * srr[c];
    }
}

// ---------------------------------------------------------------------------
// Output GEMM  out = z @ Wo^T
// ---------------------------------------------------------------------------
__global__ void gemm_out(const float* __restrict__ zin,
                         const float* __restrict__ Wo,
                         float* __restrict__ out)
{
    __shared__ float As[2][BM][LDK];
    __shared__ float Bs[2][BN][LDK];

    const int bm0 = blockIdx.x * BM;
    const int bn0 = blockIdx.y * BN;

    v8f acc[4];
    gemm_mainloop(zin, Wo, As, Bs, acc, bm0, bn0);

    const int tid  = threadIdx.x;
    const int lane = tid & 31;
    const int wave = tid >> 5;
    const int mt   = wave & 3;
    const int nb   = (wave >> 2) * 64;
    const int half = (lane >> 4) * 8;
    const int gn_b = bn0 + nb + (lane & 15);

#pragma unroll
    for (int j = 0; j < 4; ++j) {
        int gn = gn_b + j * 16;
#pragma unroll
        for (int i = 0; i < 8; ++i) {
            int gm = bm0 + mt * 16 + half + i;
            out[(size_t)gm * C_DIM + gn] = acc[j][i];
        }
    }
}

// ---------------------------------------------------------------------------
// Launch pipeline. Workspace: 3 x 96MB fp32 + inv[T] + 3 x 192KB scan states.
// d_out doubles as the rwkv temp buffer between scan and LN.
// ---------------------------------------------------------------------------
extern "C" void kernel_launch(void* const* d_in, const int* in_sizes, int n_in,
                              void* d_out, int out_size, void* d_ws, size_t ws_size,
                              hipStream_t stream)
{
    const float* x    = (const float*)d_in[0];
    const int*   sidx = (const int*)  d_in[1];
    const float* dec  = (const float*)d_in[2];
    const float* fir  = (const float*)d_in[3];
    const float* Wk   = (const float*)d_in[4];
    const float* Wv   = (const float*)d_in[5];
    const float* Wr   = (const float*)d_in[6];
    const float* Wo   = (const float*)d_in[7];
    const float* lng  = (const float*)d_in[8];
    const float* lnb  = (const float*)d_in[9];
    float* out = (float*)d_out;

    const size_t BTC = (size_t)B_DIM * T_DIM * C_DIM;
    const size_t NST = (size_t)B_DIM * NCH * C_DIM;
    float* k_s = (float*)d_ws;
    float* v_s = k_s + BTC;
    float* sr  = v_s + BTC;
    int*   inv = (int*)(sr + BTC);
    float* stA = (float*)(inv + T_DIM);
    float* stB = stA + NST;
    float* stP = stB + NST;

    build_inv<<<T_DIM / 256, 256, 0, stream>>>(sidx, inv);

    gemm_kvr<<<dim3(M_DIM / BM, C_DIM / BN, 3), 256, 0, stream>>>(
        x, Wk, Wv, Wr, k_s, v_s, sr, inv);

    wkv_chunk_local<<<(B_DIM * NCH * C_DIM) / 256, 256, 0, stream>>>(
        k_s, v_s, dec, stA, stB, stP);

    wkv_combine<<<(B_DIM * C_DIM) / 256, 256, 0, stream>>>(dec, stA, stB, stP);

    wkv_emit<<<(B_DIM * NCH * C_DIM) / 256, 256, 0, stream>>>(
        k_s, v_s, sidx, dec, fir, stA, stB, stP, out);   // rwkv -> d_out (temp)

    ln_gate<<<M_DIM, 256, 0, stream>>>(out, sr, lng, lnb, k_s);  // z -> k_s buf

    gemm_out<<<dim3(M_DIM / BM, C_DIM / BN), 256, 0, stream>>>(k_s, Wo, out);
}